// KL_LossFunc_53618371723721
// MI455X (gfx1250) — compile-verified
//
#include <hip/hip_runtime.h>
#include <math.h>

typedef __attribute__((ext_vector_type(2))) float v2f;
typedef __attribute__((ext_vector_type(8))) float v8f;

#define N_BASE   15
#define CHANNELS 4096
#define TILES    (CHANNELS / 16)   // 256 channel tiles of width 16
#define SEG      4                 // batch split for occupancy

// ---------------------------------------------------------------------------
// Stage 1: batch-sum per (base, channel) as a ones-GEMV on the matrix pipe.
// D = A(16x4, all ones) x B(4x16 of data) + C  => every D element is the
// column sum of B, i.e. the sum of the 4 batch rows loaded this chunk.
// Column sums are permutation-invariant over K, so this is robust to the
// exact B-operand VGPR layout as long as N = lane%16 (true for all CDNA5
// B layouts) and (lane-half, VGPR) <-> K is a bijection.
// One wave = one (base, 16-channel tile, batch segment). 15*256*4 = 15360 waves.
// ---------------------------------------------------------------------------
__global__ __launch_bounds__(256) void stage1_batchsum_wmma(
    const float* __restrict__ x, float* __restrict__ partial, int rowsPerSeg)
{
    const int lane = threadIdx.x & 31;
    const int wave = threadIdx.x >> 5;
    const int unit = blockIdx.x * 8 + wave;        // 0 .. 15359
    const int seg  = unit & (SEG - 1);
    const int t    = unit >> 2;
    const int tile = t & (TILES - 1);
    const int b    = t >> 8;                       // t / TILES

    const int  n0    = tile * 16;
    const int  n     = n0 + (lane & 15);
    const int  kofs  = (lane >> 4) << 1;           // 0 for lanes 0-15, 2 for 16-31
    const long ldRow = (long)N_BASE * CHANNELS;    // floats between batch rows

    // row index for batch i, base b is (i*15 + b)
    const float* p = x + ((long)(seg * rowsPerSeg + kofs) * N_BASE + b) * CHANNELS + n;

    v2f a = {1.0f, 1.0f};                          // all-ones A matrix (layout-free)
    v8f acc0 = {};
    v8f acc1 = {};

    const int iters = rowsPerSeg >> 2;             // 4 batch rows per WMMA
    for (int i = 0; i < iters; i += 2) {
        v2f b0, b1;
        b0.x = p[0];                               // rows k+0 / k+2 (per lane half)
        b0.y = p[ldRow];                           // rows k+1 / k+3
        const float* p2 = p + 4 * ldRow;
        b1.x = p2[0];                              // rows k+4 / k+6
        b1.y = p2[ldRow];                          // rows k+5 / k+7
        if (i + 2 < iters)
            __builtin_prefetch(p + 8 * ldRow, 0, 1);   // global_prefetch_b8, ~1966KB ahead
        acc0 = __builtin_amdgcn_wmma_f32_16x16x4_f32(false, a, false, b0,
                                                     (short)0, acc0, false, false);
        acc1 = __builtin_amdgcn_wmma_f32_16x16x4_f32(false, a, false, b1,
                                                     (short)0, acc1, false, false);
        p += 8 * ldRow;
    }

    v8f accs = acc0 + acc1;
    float s = accs[0];                             // D[*][lane%16] = column sum (all rows equal)
    if (lane < 16)
        partial[((long)seg * N_BASE + b) * CHANNELS + n0 + lane] = s;
}

// ---------------------------------------------------------------------------
// Stage 2: per base, merge segment partials -> per-channel mean m[c], then
// fixed-order reduce sum(m), sum(m^2) over 4096 channels -> sub_mean, sub_std.
// ---------------------------------------------------------------------------
__global__ __launch_bounds__(256) void stage2_channel_stats(
    const float* __restrict__ partial, float* __restrict__ stats, float invBatch)
{
    __shared__ float red0[8];
    __shared__ float red1[8];
    const int b   = blockIdx.x;
    const int tid = threadIdx.x;

    float sm = 0.0f, sm2 = 0.0f;
    for (int c = tid; c < CHANNELS; c += 256) {
        float S = 0.0f;
        for (int seg = 0; seg < SEG; ++seg)
            S += partial[((long)seg * N_BASE + b) * CHANNELS + c];
        float m = S * invBatch;
        sm  += m;
        sm2 += m * m;
    }
    // wave32 tree reduction
    for (int off = 16; off >= 1; off >>= 1) {
        sm  += __shfl_xor(sm,  off, 32);
        sm2 += __shfl_xor(sm2, off, 32);
    }
    const int lane = tid & 31, wv = tid >> 5;
    if (lane == 0) { red0[wv] = sm; red1[wv] = sm2; }
    __syncthreads();
    if (tid == 0) {
        float tsm = 0.0f, tsm2 = 0.0f;
        for (int w = 0; w < 8; ++w) { tsm += red0[w]; tsm2 += red1[w]; }
        float meanb = tsm / (float)CHANNELS;
        float var   = (tsm2 - tsm * tsm / (float)CHANNELS) / (float)(CHANNELS - 1);
        if (var < 0.0f) var = 0.0f;
        stats[b]          = meanb;        // sub_mean[b]
        stats[N_BASE + b] = sqrtf(var);   // sub_std[b] (ddof=1)
    }
}

// ---------------------------------------------------------------------------
// Stage 3: tiny group/pairwise epilogue on one thread (double precision).
// SAME_IDS = [[0,1,2],[3,4],[5],[6,7,8,9],[10,11],[12,13,14]], MEAN_STD = 1.0
// ---------------------------------------------------------------------------
__global__ void stage3_loss(const float* __restrict__ stats, float* __restrict__ out)
{
    if (threadIdx.x != 0 || blockIdx.x != 0) return;
    const int gsize[6]  = {3, 2, 1, 4, 2, 3};
    const int gstart[6] = {0, 3, 5, 6, 10, 12};

    double ss[6], meanv[6], msum[6];
    for (int g = 0; g < 6; ++g) {
        int n = gsize[g], s0 = gstart[g];
        if (n == 1) {
            ss[g]    = (double)stats[N_BASE + s0];
            meanv[g] = (double)stats[s0];
            msum[g]  = (double)stats[s0];
        } else {
            double mu_s = 0.0, mu_m = 0.0;
            for (int k = 0; k < n; ++k) {
                mu_s += (double)stats[N_BASE + s0 + k];
                mu_m += (double)stats[s0 + k];
            }
            mu_s /= n; mu_m /= n;
            double vs = 0.0, vm = 0.0;
            for (int k = 0; k < n; ++k) {
                double ds = (double)stats[N_BASE + s0 + k] - mu_s; vs += ds * ds;
                double dm = (double)stats[s0 + k]          - mu_m; vm += dm * dm;
            }
            ss[g]    = sqrt(vs / (n - 1));    // std of group stds (ddof=1)
            msum[g]  = sqrt(vm / (n - 1));    // std of group means (ddof=1)
            meanv[g] = mu_m;                  // mean of group means
        }
    }
    double loss = 0.0;
    for (int i = 0; i < 6; ++i) {
        double sub = 5e-5;
        for (int j = i + 1; j < 6; ++j) {
            double d1 = ss[i] - ss[j];
            double d2 = meanv[i] - meanv[j];
            sub += exp(d1 * d1) + exp(d2 * d2);   // MEAN_STD = 1.0
        }
        loss += log(1.0 + msum[i] / (sub + msum[i]));
    }
    out[0] = (float)loss;
}

// ---------------------------------------------------------------------------
extern "C" void kernel_launch(void* const* d_in, const int* in_sizes, int n_in,
                              void* d_out, int out_size, void* d_ws, size_t ws_size,
                              hipStream_t stream) {
    const float* meta1 = (const float*)d_in[0];
    float* out = (float*)d_out;
    float* ws  = (float*)d_ws;

    const long total = (long)in_sizes[0];
    const int  batch = (int)(total / ((long)N_BASE * CHANNELS));  // 512
    const int  rowsPerSeg = batch / SEG;                          // 128

    float* partial = ws;                                          // SEG*15*4096 f32
    float* stats   = ws + (long)SEG * N_BASE * CHANNELS;          // 30 f32

    const int units   = N_BASE * TILES * SEG;                     // 15360 waves
    const int blocks1 = units / 8;                                // 1920 blocks x 256 thr

    stage1_batchsum_wmma<<<blocks1, 256, 0, stream>>>(meta1, partial, rowsPerSeg);
    stage2_channel_stats<<<N_BASE, 256, 0, stream>>>(partial, stats, 1.0f / (float)batch);
    stage3_loss<<<1, 32, 0, stream>>>(stats, out);
}